// LossWrapper_40733469835867
// MI455X (gfx1250) — compile-verified
//
#include <hip/hip_runtime.h>
#include <hip/hip_bf16.h>
#include <math.h>
#include <stdint.h>

typedef __attribute__((ext_vector_type(2))) float v2f;
typedef __attribute__((ext_vector_type(8))) float v8f;

#define NCLS 200000
#define DIM 192
#define ROWS 128            // B*M
#define WSTRIDE 196         // LDS row stride in floats (196 % 64 == 4 -> conflict free)
#define COLS_PER_BLOCK 256
#define NTILES (COLS_PER_BLOCK / 16)
#define KSTEPS 48           // 192 / 4
#define SCALE_F 30.0f
#define COS_M_F 0.8775825618903728f
#define SIN_M_F 0.479425538604203f
#define TH_F   (-0.8775825618903728f)
#define MM_F    0.2397127693021015f

__constant__ int PERMS24[24][4] = {
  {0,1,2,3},{0,1,3,2},{0,2,1,3},{0,2,3,1},{0,3,1,2},{0,3,2,1},
  {1,0,2,3},{1,0,3,2},{1,2,0,3},{1,2,3,0},{1,3,0,2},{1,3,2,0},
  {2,0,1,3},{2,0,3,1},{2,1,0,3},{2,1,3,0},{2,3,0,1},{2,3,1,0},
  {3,0,1,2},{3,0,2,1},{3,1,0,2},{3,1,2,0},{3,2,0,1},{3,2,1,0}
};

// async copy of one thread's 48-byte chunk of a weight tile into LDS.
// INST_OFFSET is added to BOTH the LDS and global address (ISA 15.18.3),
// so one (ldsaddr, gaddr) pair covers all three B128 transfers.
__device__ __forceinline__ void issue_async48(uint32_t ldsa, const float* gp) {
  asm volatile(
      "global_load_async_to_lds_b128 %0, %1, off\n\t"
      "global_load_async_to_lds_b128 %0, %1, off offset:16\n\t"
      "global_load_async_to_lds_b128 %0, %1, off offset:32"
      :: "v"(ldsa), "v"(gp) : "memory");
}

__device__ __forceinline__ void wait_async0() {
  asm volatile("s_wait_asynccnt 0x0" ::: "memory");
}

// ---------------- normalize pred_embs rows (128 x 192) ----------------
__global__ __launch_bounds__(192) void k_norm_x(const float* __restrict__ pe,
                                                float* __restrict__ xnorm) {
  int row = blockIdx.x;
  int t = threadIdx.x;
  float v = pe[(size_t)row * DIM + t];
  float ss = v * v;
  for (int m = 1; m < 32; m <<= 1) ss += __shfl_xor(ss, m);
  __shared__ float red[6];
  __shared__ float sinv;
  if ((t & 31) == 0) red[t >> 5] = ss;
  __syncthreads();
  if (t == 0) {
    float s = 0.f;
    for (int i = 0; i < 6; ++i) s += red[i];
    sinv = 1.0f / fmaxf(sqrtf(s), 1e-12f);
  }
  __syncthreads();
  xnorm[(size_t)row * DIM + t] = v * sinv;
}

// ---------------- gather the 4 label indices per batch row ----------------
__global__ __launch_bounds__(256) void k_labels(const int* __restrict__ gt,
                                                int* __restrict__ labels) {
  __shared__ int cnt;
  int b = blockIdx.x;
  if (threadIdx.x == 0) cnt = 0;
  __syncthreads();
  for (int c = threadIdx.x; c < NCLS; c += blockDim.x) {
    if (gt[(size_t)b * NCLS + c] == 1) {
      int k = atomicAdd(&cnt, 1);
      if (k < 4) labels[b * 4 + k] = c;   // order irrelevant: perm-matching is set-invariant
    }
  }
}

// ---------------- main WMMA GEMM + shifted sum-exp partials ----------------
__global__ __launch_bounds__(256) void k_gemm_sumexp(const float* __restrict__ xnorm,
                                                     const float* __restrict__ weight,
                                                     float* __restrict__ partial) {
  __shared__ float lw[2][16 * WSTRIDE];   // double-buffered weight tile
  __shared__ float sinv[16];

  int tid  = threadIdx.x;
  int wave = tid >> 5;
  int lane = tid & 31;
  int j  = lane & 15;     // A-row within tile / B-col within tile (ISA 7.12.2)
  int hi = lane >> 4;     // lane-half -> K offset 2*hi, D-row offset 8*hi
  int colBase = blockIdx.x * COLS_PER_BLOCK;

  // staging geometry: each thread owns 12 consecutive floats of one tile row
  int srow = tid >> 4;            // class row within tile, 0..15
  int scol = (tid & 15) * 12;     // 16-byte aligned (48B chunks)

  // A fragments for this wave's 16 rows, all 48 K-steps, kept in VGPRs.
  v2f afrag[KSTEPS];
  {
    const float* xr = xnorm + (size_t)(wave * 16 + j) * DIM + 2 * hi;
#pragma unroll
    for (int kt = 0; kt < KSTEPS; ++kt) {
      afrag[kt].x = xr[4 * kt + 0];
      afrag[kt].y = xr[4 * kt + 1];
    }
  }

  // prologue: kick off tile 0 into buffer 0
  {
    int g = colBase + srow;
    if (g < NCLS)
      issue_async48((uint32_t)(uintptr_t)&lw[0][srow * WSTRIDE + scol],
                    weight + (size_t)g * DIM + scol);
  }

  float acc[8];
#pragma unroll
  for (int r = 0; r < 8; ++r) acc[r] = 0.f;

  for (int ct = 0; ct < NTILES; ++ct) {
    int buf = ct & 1;

    wait_async0();        // my async writes for tile ct are done
    __syncthreads();      // everyone's writes visible; prev reads of buf^1 done

    // start fetching tile ct+1 into the other buffer (overlaps with compute)
    if (ct + 1 < NTILES) {
      int g = colBase + (ct + 1) * 16 + srow;
      if (g < NCLS)
        issue_async48((uint32_t)(uintptr_t)&lw[buf ^ 1][srow * WSTRIDE + scol],
                      weight + (size_t)g * DIM + scol);
    }

    // per-class inverse norms (16 lanes per row, shuffle-reduce)
    {
      int r = tid >> 4, sub = tid & 15;
      float ss = 0.f;
      for (int c = sub; c < DIM; c += 16) { float v = lw[buf][r * WSTRIDE + c]; ss += v * v; }
      ss += __shfl_xor(ss, 1); ss += __shfl_xor(ss, 2);
      ss += __shfl_xor(ss, 4); ss += __shfl_xor(ss, 8);
      if (sub == 0) sinv[r] = 1.0f / fmaxf(sqrtf(ss), 1e-12f);
    }
    __syncthreads();

    // pure ds_load + wmma inner loop (norm scaling folded into epilogue)
    v8f cacc;
#pragma unroll
    for (int r = 0; r < 8; ++r) cacc[r] = 0.f;

#pragma unroll
    for (int kt = 0; kt < KSTEPS; ++kt) {
      const float* bp = &lw[buf][j * WSTRIDE + 4 * kt + 2 * hi];
      v2f bfrag;
      bfrag.x = bp[0];
      bfrag.y = bp[1];
      cacc = __builtin_amdgcn_wmma_f32_16x16x4_f32(
          false, afrag[kt], false, bfrag, (short)0, cacc, false, false);
    }

    // epilogue: scale by 1/||W_col|| and accumulate exp(30*cos - 30)
    int col = colBase + ct * 16 + j;
    if (col < NCLS) {
      float invn = sinv[j];
#pragma unroll
      for (int r = 0; r < 8; ++r)
        acc[r] += __expf(SCALE_F * (cacc[r] * invn) - SCALE_F);
    }
  }

  // reduce each row's 16 columns across the lane-half, write block partial
#pragma unroll
  for (int r = 0; r < 8; ++r) {
    float s = acc[r];
    s += __shfl_xor(s, 1); s += __shfl_xor(s, 2);
    s += __shfl_xor(s, 4); s += __shfl_xor(s, 8);
    if (j == 0)
      partial[(size_t)blockIdx.x * ROWS + wave * 16 + r + 8 * hi] = s;
  }
}

// ---------------- final fuse: lse, phi, perm-min, means ----------------
__global__ __launch_bounds__(256) void k_final(const float* __restrict__ xnorm,
                                               const float* __restrict__ weight,
                                               const float* __restrict__ pred_ps,
                                               const int* __restrict__ labels,
                                               const float* __restrict__ partial,
                                               int nblk, float* __restrict__ out) {
  __shared__ float sumf[ROWS];
  __shared__ float cosl[512];   // [b][m][s]
  __shared__ float Cmat[512];   // [b][s][m]
  __shared__ float lb[32 * 4];
  int t = threadIdx.x;

  if (t < ROWS) {
    double s = 0.0;
    for (int k = 0; k < nblk; ++k) s += (double)partial[(size_t)k * ROWS + t];
    sumf[t] = (float)s;
  }

  for (int task = t; task < 512; task += 256) {
    int b = task >> 4, m = (task >> 2) & 3, s = task & 3;
    int lab = labels[b * 4 + s];
    const float* xr = xnorm + (size_t)(b * 4 + m) * DIM;
    const float* wr = weight + (size_t)lab * DIM;
    float dot = 0.f, ss = 0.f;
    for (int c = 0; c < DIM; ++c) { float w = wr[c]; dot += xr[c] * w; ss += w * w; }
    cosl[task] = dot / fmaxf(sqrtf(ss), 1e-12f);
  }
  __syncthreads();

  for (int task = t; task < 512; task += 256) {
    int b = task >> 4, m = (task >> 2) & 3, s = task & 3;
    float cl = cosl[task];
    float sl = sqrtf(fminf(fmaxf(1.0f - cl * cl, 0.f), 1.f));
    float phi = cl * COS_M_F - sl * SIN_M_F;
    phi = (cl > TH_F) ? phi : (cl - MM_F);
    float adj = sumf[b * 4 + m] - expf(SCALE_F * cl - SCALE_F) + expf(SCALE_F * phi - SCALE_F);
    float ce = SCALE_F + logf(adj) - SCALE_F * phi;   // == mx + log(adj) - SCALE*phi
    Cmat[b * 16 + s * 4 + m] = ce;
  }
  __syncthreads();

  if (t < 32) {
    int b = t;
    float best = 3.4e38f;
    for (int p = 0; p < 24; ++p) {
      float pc = 0.f;
      for (int s = 0; s < 4; ++s) pc += Cmat[b * 16 + s * 4 + PERMS24[p][s]];
      best = fminf(best, pc);
    }
    float Lspk = best * 0.25f;
    float le = 0.f;
    for (int m = 0; m < 4; ++m) {   // S==M -> t_exist is all ones
      float pv = fminf(fmaxf(pred_ps[b * 4 + m], 1e-6f), 1.0f - 1e-6f);
      le += logf(pv);
    }
    float Lexist = -le * 0.25f;
    float p3 = fminf(fmaxf(pred_ps[b * 4 + 3], 1e-6f), 1.0f - 1e-6f);
    float Lstop = -logf(p3);
    float Ltot = 0.01f * Lspk + 2.5f * Lexist + 5.0f * Lstop;
    lb[b * 4 + 0] = Ltot; lb[b * 4 + 1] = Lspk;
    lb[b * 4 + 2] = Lexist; lb[b * 4 + 3] = Lstop;
  }
  __syncthreads();

  if (t == 0) {
    float a0 = 0, a1 = 0, a2 = 0, a3 = 0;
    for (int b = 0; b < 32; ++b) {
      a0 += lb[b * 4 + 0]; a1 += lb[b * 4 + 1];
      a2 += lb[b * 4 + 2]; a3 += lb[b * 4 + 3];
    }
    out[0] = a0 / 32.f; out[1] = a1 / 32.f;
    out[2] = a2 / 32.f; out[3] = a3 / 32.f;
  }
}

extern "C" void kernel_launch(void* const* d_in, const int* in_sizes, int n_in,
                              void* d_out, int out_size, void* d_ws, size_t ws_size,
                              hipStream_t stream) {
  const float* pred_embs = (const float*)d_in[0];   // (32,4,192) f32
  const float* pred_ps   = (const float*)d_in[1];   // (32,4) f32
  const int*   gt_labels = (const int*)d_in[2];     // (32,200000) i32
  const float* weight    = (const float*)d_in[3];   // (200000,192) f32
  float* out = (float*)d_out;

  const int NBLK = (NCLS + COLS_PER_BLOCK - 1) / COLS_PER_BLOCK;  // 782

  char* ws = (char*)d_ws;
  float* xnorm   = (float*)ws;                                       // 128*192*4 = 98304 B
  float* partial = (float*)(ws + (size_t)ROWS * DIM * 4);            // NBLK*128*4 ≈ 400 KB
  int*   labels  = (int*)(ws + (size_t)ROWS * DIM * 4 + (size_t)NBLK * ROWS * 4);

  k_norm_x<<<ROWS, DIM, 0, stream>>>(pred_embs, xnorm);
  k_labels<<<32, 256, 0, stream>>>(gt_labels, labels);
  k_gemm_sumexp<<<NBLK, 256, 0, stream>>>(xnorm, weight, partial);
  k_final<<<1, 256, 0, stream>>>(xnorm, weight, pred_ps, labels, partial, NBLK, out);
}